// NUFFTLayer_59906203845041
// MI455X (gfx1250) — compile-verified
//
#include <hip/hip_runtime.h>
#include <math.h>

#define MGRID 4097
#define KPAD  4100           // K padded to multiple of 4 (cols 4097..4099 zero)
#define NPART 2048
#define BATCH 8
#define PI_D  3.14159265358979323846

// workspace layout (float offsets), ws assumed >= 256B aligned
#define WS_HR   0                       // MGRID+8 padded circular kernel
#define WS_RED  4160                    // 16 * KPAD  (rows 8..15 zero)
#define WS_IRF  (4160 + 16 * KPAD)     // 8 * MGRID convolved grid

typedef float v2f __attribute__((ext_vector_type(2)));
typedef float v8f __attribute__((ext_vector_type(8)));

// ---------------- Kernel 1: circular kernel C[d] = sum_k W(k) cos(2*pi*k*d/M) ----
// W(k) = amplitud*4pi/(k^2+(5*shift)^2) * (pi/tau) * exp(2*tau*k^2), even in k.
// Cnorm[d] = C[d] / M^3 folds all normalizations (fft/SCALE, ifft, /SCALE).
__global__ __launch_bounds__(256) void build_kernel(const float* __restrict__ shift,
                                                    const float* __restrict__ amplitud,
                                                    float* __restrict__ ws) {
    __shared__ double part[256];
    const int d   = blockIdx.x;           // 0..MGRID-1
    const int tid = threadIdx.x;

    const double tau   = 12.0 / ((double)MGRID * (double)MGRID);
    const float  s5    = 5.0f * shift[0];
    const float  s5sq  = s5 * s5;
    const float  amp4pi = amplitud[0] * (float)(4.0 * PI_D);
    const float  pi_tau = (float)(PI_D / tau);
    const float  twotau = (float)(2.0 * tau);
    const float  twopim = (float)(2.0 * PI_D / (double)MGRID);

    double acc = 0.0;
    for (int k = 1 + tid; k <= MGRID / 2; k += 256) {
        float kf = (float)k;
        float k2 = kf * kf;
        float w  = (amp4pi / (k2 + s5sq)) * pi_tau * expf(twotau * k2);
        unsigned r = ((unsigned)k * (unsigned)d) % (unsigned)MGRID;  // keep phase in [0,2pi)
        acc += (double)(w * cosf(twopim * (float)r));
    }
    part[tid] = acc;
    __syncthreads();
    for (int s = 128; s > 0; s >>= 1) {           // deterministic tree reduce
        if (tid < s) part[tid] += part[tid + s];
        __syncthreads();
    }
    if (tid == 0) {
        double w0 = (double)((amp4pi / s5sq) * pi_tau);          // k = 0 term
        double C  = w0 + 2.0 * part[0];
        double invM3 = 1.0 / ((double)MGRID * (double)MGRID * (double)MGRID);
        float val = (float)(C * invM3);
        ws[WS_HR + d] = val;
        if (d < 8) ws[WS_HR + MGRID + d] = val;   // circular pad for b64-style pairs
    }
}

// ---------------- Kernel 2: spread particles -> red[16][KPAD] (zero padded) ------
__global__ __launch_bounds__(256) void spread_kernel(const float* __restrict__ x,
                                                     float* __restrict__ ws) {
    __shared__ float xs[NPART];
    const int tid  = threadIdx.x;
    const int row  = blockIdx.x / 17;     // 0..15 (rows 8..15 = zero pad)
    const int j    = (blockIdx.x % 17) * 256 + tid;
    float* red = ws + WS_RED;
    if (row < BATCH) {
        for (int i = tid; i < NPART; i += 256) xs[i] = x[row * NPART + i];
    }
    __syncthreads();
    if (j >= KPAD) return;
    float val = 0.0f;
    if (row < BATCH && j < MGRID) {
        const float inv4tau = (float)(((double)MGRID * (double)MGRID) / 48.0); // 1/(4*tau)
        const float Lf = (float)(2.0 * PI_D);
        const float xg = (float)(-PI_D + (double)j * (2.0 * PI_D / (double)MGRID));
        float acc = 0.0f;
        for (int n = 0; n < NPART; ++n) {         // fixed order -> deterministic
            float dd = xs[n] - xg;
            float dm = dd - Lf, dp = dd + Lf;
            acc += expf(-dd * dd * inv4tau) + expf(-dm * dm * inv4tau)
                 + expf(-dp * dp * inv4tau);
        }
        val = acc;
    }
    red[row * KPAD + j] = val;
}

// ---------------- Kernel 3: circular convolution as fp32 WMMA matmul -------------
// irfft[b, m] = sum_j red[b, j] * Cnorm[(m-j) mod M], Cnorm even -> gather hr[(j-m) mod M].
// One wave per 16-column output tile; D = A(16x4)*B(4x16)+D over 1025 K-steps.
// 32-bit operand layout (ISA 7.12.2): lanes 0-15 = first K/M half, 16-31 = second.
__global__ __launch_bounds__(256) void conv_wmma_kernel(float* __restrict__ ws) {
    __shared__ float hr[MGRID + 15];
    const int tid = threadIdx.x;
    {
        const float* hrg = ws + WS_HR;
        for (int i = tid; i < MGRID + 8; i += 256) hr[i] = hrg[i];
    }
    __syncthreads();

    const int lane = tid & 31;
    const int r    = lane & 15;           // A row / B,D column within tile
    const int hh   = lane >> 4;           // half-wave: K-pair select
    const int tile = blockIdx.x * 8 + (tid >> 5);
    const int m0   = tile * 16;
    if (m0 >= MGRID) return;              // wave-uniform: EXEC stays all-ones for WMMA

    const float* redrow = ws + WS_RED + r * KPAD;   // rows 8..15 are zero (batch pad)
    int t = ((2 * hh - m0 - r) % MGRID + MGRID) % MGRID;   // (j + 2hh - m0 - r) mod M at j=0

    v8f acc = {0.f, 0.f, 0.f, 0.f, 0.f, 0.f, 0.f, 0.f};
    #pragma unroll 4
    for (int j = 0; j < KPAD; j += 4) {
        v2f a = *(const v2f*)(redrow + j + 2 * hh);        // A[r][j+2hh], A[r][j+2hh+1]
        v2f b;
        b.x = hr[t];                                       // B[2hh  ][r]
        b.y = hr[t + 1];                                   // B[2hh+1][r] (pad covers wrap)
        acc = __builtin_amdgcn_wmma_f32_16x16x4_f32(false, a, false, b,
                                                    (short)0, acc, false, false);
        t += 4;
        if (t >= MGRID) t -= MGRID;
    }
    if (hh == 0) {                        // VGPR i holds D[row=i][col=r] for lanes 0-15
        int m = m0 + r;
        if (m < MGRID) {
            float* irf = ws + WS_IRF;
            #pragma unroll
            for (int i = 0; i < BATCH; ++i) irf[i * MGRID + m] = acc[i];
        }
    }
}

// ---------------- Kernel 4: interpolate back to particles ------------------------
__global__ __launch_bounds__(256) void interp_kernel(const float* __restrict__ x,
                                                     const float* __restrict__ ws,
                                                     float* __restrict__ out) {
    __shared__ float irow[MGRID];
    const int tid = threadIdx.x;
    const int b   = blockIdx.x >> 3;
    const int n   = ((blockIdx.x & 7) << 8) + tid;        // 0..2047 exactly
    {
        const float* irf = ws + WS_IRF + b * MGRID;
        for (int i = tid; i < MGRID; i += 256) irow[i] = irf[i];
    }
    __syncthreads();
    const float xv = x[b * NPART + n];
    const float inv4tau = (float)(((double)MGRID * (double)MGRID) / 48.0);
    const float Lf  = (float)(2.0 * PI_D);
    const float xlo = (float)(-PI_D);
    const float hstep = (float)(2.0 * PI_D / (double)MGRID);
    float acc = 0.0f;
    for (int m = 0; m < MGRID; ++m) {
        float xg = fmaf((float)m, hstep, xlo);
        float dd = xv - xg;
        float dm = dd - Lf, dp = dd + Lf;
        float g = expf(-dd * dd * inv4tau) + expf(-dm * dm * inv4tau)
                + expf(-dp * dp * inv4tau);
        acc = fmaf(irow[m], g, acc);
    }
    out[b * NPART + n] = acc;
}

extern "C" void kernel_launch(void* const* d_in, const int* in_sizes, int n_in,
                              void* d_out, int out_size, void* d_ws, size_t ws_size,
                              hipStream_t stream) {
    (void)in_sizes; (void)n_in; (void)out_size; (void)ws_size;
    const float* x        = (const float*)d_in[0];
    const float* shift    = (const float*)d_in[1];
    const float* amplitud = (const float*)d_in[2];
    float* ws  = (float*)d_ws;
    float* out = (float*)d_out;

    build_kernel <<<MGRID,   256, 0, stream>>>(shift, amplitud, ws);
    spread_kernel<<<16 * 17, 256, 0, stream>>>(x, ws);
    conv_wmma_kernel<<<33,   256, 0, stream>>>(ws);     // 33*8 waves >= 257 tiles
    interp_kernel<<<64,      256, 0, stream>>>(x, ws, out);
}